// Gate_48722109006497
// MI455X (gfx1250) — compile-verified
//
#include <hip/hip_runtime.h>
#include <hip/hip_bf16.h>

typedef float v2f __attribute__((ext_vector_type(2)));
typedef float v8f __attribute__((ext_vector_type(8)));

#define GATE_N_TOKENS   16384
#define GATE_DIM        2048
#define GATE_N_EXPERTS  64
#define GATE_TOPK       8
#define GATE_ROUTE_SCALE 1.0f

// ---------------------------------------------------------------------------
// Phase 1: logits = x @ W^T via V_WMMA_F32_16X16X4_F32.
// Block = 256 threads = 8 waves. Each wave computes a 32-token x 64-expert
// tile (8 C-tiles of v8f: 2 token tiles x 4 expert tiles), looping K=2048 in
// steps of 4. Per K-step: 2 A-fragment loads + 4 B-fragment loads feed
// 8 WMMAs (0.75 loads/WMMA), amortizing W traffic across two token tiles.
//
// f32 16x16x4 fragment layout (wave32):
//   A (16x4, MxK): lane<16 -> M=lane, K={k,k+1}; lane>=16 -> M=lane-16, K={k+2,k+3}
//   B (4x16, KxN): lane<16 -> N=lane, K={k,k+1}; lane>=16 -> N=lane-16, K={k+2,k+3}
// Both are a contiguous per-lane float2 at row*DIM + k + 2*(lane>>4).
//   C/D (16x16): VGPR r -> M=r (lanes 0-15), M=r+8 (lanes 16-31); N=lane&15.
// ---------------------------------------------------------------------------
__global__ __launch_bounds__(256) void gate_logits_wmma(
    const float* __restrict__ x, const float* __restrict__ W,
    float* __restrict__ scores) {
  const int wave = threadIdx.x >> 5;
  const int lane = threadIdx.x & 31;
  const int mt   = (blockIdx.x * 8 + wave) * 32;   // token-tile base (32 tokens)
  const int row  = lane & 15;
  const int koff = (lane >> 4) << 1;               // 0 or 2

  const float* ap0 = x + (size_t)(mt +  0 + row) * GATE_DIM + koff;
  const float* ap1 = x + (size_t)(mt + 16 + row) * GATE_DIM + koff;
  const float* b0  = W + (size_t)( 0 + row) * GATE_DIM + koff;
  const float* b1  = W + (size_t)(16 + row) * GATE_DIM + koff;
  const float* b2  = W + (size_t)(32 + row) * GATE_DIM + koff;
  const float* b3  = W + (size_t)(48 + row) * GATE_DIM + koff;

  v8f c00 = {}, c10 = {}, c20 = {}, c30 = {};   // token tile 0, expert tiles 0..3
  v8f c01 = {}, c11 = {}, c21 = {}, c31 = {};   // token tile 1, expert tiles 0..3

#pragma unroll 2
  for (int k = 0; k < GATE_DIM; k += 4) {
    v2f a0 = *(const v2f*)(ap0 + k);
    v2f a1 = *(const v2f*)(ap1 + k);
    v2f w0 = *(const v2f*)(b0 + k);
    v2f w1 = *(const v2f*)(b1 + k);
    v2f w2 = *(const v2f*)(b2 + k);
    v2f w3 = *(const v2f*)(b3 + k);
    c00 = __builtin_amdgcn_wmma_f32_16x16x4_f32(false, a0, false, w0, (short)0, c00, false, false);
    c01 = __builtin_amdgcn_wmma_f32_16x16x4_f32(false, a1, false, w0, (short)0, c01, false, false);
    c10 = __builtin_amdgcn_wmma_f32_16x16x4_f32(false, a0, false, w1, (short)0, c10, false, false);
    c11 = __builtin_amdgcn_wmma_f32_16x16x4_f32(false, a1, false, w1, (short)0, c11, false, false);
    c20 = __builtin_amdgcn_wmma_f32_16x16x4_f32(false, a0, false, w2, (short)0, c20, false, false);
    c21 = __builtin_amdgcn_wmma_f32_16x16x4_f32(false, a1, false, w2, (short)0, c21, false, false);
    c30 = __builtin_amdgcn_wmma_f32_16x16x4_f32(false, a0, false, w3, (short)0, c30, false, false);
    c31 = __builtin_amdgcn_wmma_f32_16x16x4_f32(false, a1, false, w3, (short)0, c31, false, false);
  }

  // Scatter C tiles to scores[token][expert]. Stores coalesce across lanes
  // (16 lanes x 4B contiguous per expert tile).
  const int tokHi = (lane >> 4) << 3;  // 0 or 8
  const int n     = lane & 15;
#pragma unroll
  for (int r = 0; r < 8; ++r) {
    {
      const int token = mt + r + tokHi;            // token tile 0
      float* dst = scores + (size_t)token * GATE_N_EXPERTS;
      dst[ 0 + n] = c00[r];
      dst[16 + n] = c10[r];
      dst[32 + n] = c20[r];
      dst[48 + n] = c30[r];
    }
    {
      const int token = mt + 16 + r + tokHi;       // token tile 1
      float* dst = scores + (size_t)token * GATE_N_EXPERTS;
      dst[ 0 + n] = c01[r];
      dst[16 + n] = c11[r];
      dst[32 + n] = c21[r];
      dst[48 + n] = c31[r];
    }
  }
}

// ---------------------------------------------------------------------------
// Phase 2: per-token softmax over 64 experts + top-8 select.
// One wave per token; each lane owns 2 experts. 8 waves per block.
// ---------------------------------------------------------------------------
__global__ __launch_bounds__(256) void gate_softmax_topk(
    const float* __restrict__ scores, float* __restrict__ w_out,
    int* __restrict__ i_out) {
  const int wave  = threadIdx.x >> 5;
  const int lane  = threadIdx.x & 31;
  const int token = blockIdx.x * 8 + wave;

  v2f s = *(const v2f*)(scores + (size_t)token * GATE_N_EXPERTS + lane * 2);

  // Row max (32-lane reduction).
  float m = fmaxf(s.x, s.y);
#pragma unroll
  for (int off = 16; off >= 1; off >>= 1)
    m = fmaxf(m, __shfl_xor(m, off, 32));

  float e0 = expf(s.x - m);
  float e1 = expf(s.y - m);

  float sum = e0 + e1;
#pragma unroll
  for (int off = 16; off >= 1; off >>= 1)
    sum += __shfl_xor(sum, off, 32);

  const float inv = 1.0f / sum;
  float v0 = e0 * inv, v1 = e1 * inv;
  const int i0 = lane * 2, i1 = lane * 2 + 1;

  // Iterative wave-wide argmax, tie -> lower expert index (matches top_k).
#pragma unroll
  for (int j = 0; j < GATE_TOPK; ++j) {
    float bv; int bi;
    if (v0 > v1 || (v0 == v1 && i0 < i1)) { bv = v0; bi = i0; }
    else                                  { bv = v1; bi = i1; }
#pragma unroll
    for (int off = 16; off >= 1; off >>= 1) {
      float ov = __shfl_xor(bv, off, 32);
      int   oi = __shfl_xor(bi, off, 32);
      if (ov > bv || (ov == bv && oi < bi)) { bv = ov; bi = oi; }
    }
    if (lane == 0) {
      w_out[(size_t)token * GATE_TOPK + j] = bv * GATE_ROUTE_SCALE;
      i_out[(size_t)token * GATE_TOPK + j] = bi;
    }
    if (i0 == bi) v0 = -1.0f;  // probs are > 0, so -1 is below everything
    if (i1 == bi) v1 = -1.0f;
  }
}

extern "C" void kernel_launch(void* const* d_in, const int* in_sizes, int n_in,
                              void* d_out, int out_size, void* d_ws, size_t ws_size,
                              hipStream_t stream) {
  const float* x = (const float*)d_in[0];  // [16384, 2048] fp32
  const float* W = (const float*)d_in[1];  // [64, 2048] fp32

  float* scores = (float*)d_ws;            // [16384, 64] fp32 scratch (4 MB)

  float* w_out = (float*)d_out;                        // weights [16384, 8] fp32
  int*   i_out = (int*)((float*)d_out + (size_t)GATE_N_TOKENS * GATE_TOPK);  // indices int32

  // Phase 1: 32 tokens/wave * 8 waves/block = 256 tokens/block.
  dim3 grid1(GATE_N_TOKENS / 256), block1(256);
  gate_logits_wmma<<<grid1, block1, 0, stream>>>(x, W, scores);

  // Phase 2: 1 token/wave * 8 waves/block.
  dim3 grid2(GATE_N_TOKENS / 8), block2(256);
  gate_softmax_topk<<<grid2, block2, 0, stream>>>(scores, w_out, i_out);
}